// BinaryHead_489626272200
// MI455X (gfx1250) — compile-verified
//
#include <hip/hip_runtime.h>

typedef float v2f __attribute__((ext_vector_type(2)));
typedef float v8f __attribute__((ext_vector_type(8)));

namespace {
constexpr int   kEmb             = 2048;
constexpr float kScale           = 16.0f;
constexpr int   kRowsPerWave     = 16;
constexpr int   kWavesPerBlock   = 8;     // 256 threads = 8 wave32
constexpr int   kThreadsPerBlock = kWavesPerBlock * 32;
}

// One wave computes a 16-row tile of the output:
//   D(16x16) = A(W, 16x4-chunk) x B(fea, 4x16-chunk) accumulated over K=2048,
// plus per-row sum-of-squares in the same single streaming pass over fea.
// logit[r][c] = (dot(fea[r], W[c]) / ||fea[r]|| + b[c]) * S
__global__ __launch_bounds__(kThreadsPerBlock)
void binary_head_wmma_f32(const float* __restrict__ fea,
                          const float* __restrict__ W,
                          const float* __restrict__ bias,
                          float* __restrict__ out,
                          int numRows)
{
    const int lane = threadIdx.x & 31;
    const int wave = threadIdx.x >> 5;
    const int tile = blockIdx.x * kWavesPerBlock + wave;
    const int row0 = tile * kRowsPerWave;
    if (row0 >= numRows) return;   // wave-uniform; EXEC stays all-1s for WMMA

    const int n = lane & 15;   // B: fea row within tile / A: class index (m)
    const int h = lane >> 4;   // half selects K offset {2h, 2h+1} within K=4 step

    const float* feaP = fea + (size_t)(row0 + n) * kEmb + (h << 1);
    const float* wP   = W   + (size_t)(n & 3)    * kEmb + (h << 1);
    const float  keepA = (n < 4) ? 1.0f : 0.0f;  // zero A rows 4..15 (branch-free)

    v8f acc0 = {};
    v8f acc1 = {};
    v2f ss2  = {};   // packed sum-of-squares accumulator -> v_pk_fma_f32

#pragma unroll 8
    for (int k = 0; k < kEmb; k += 8) {
        // B fragments: fea — 128 MB streamed once; NT hint so it does not
        // evict the L2-resident W / hot lines (L2 = 192 MB, fea = 128 MB).
        v2f b0 = __builtin_nontemporal_load((const v2f*)(feaP + k));
        v2f b1 = __builtin_nontemporal_load((const v2f*)(feaP + k + 4));
        // A fragments: W, tiny and reused by every wave -> regular temporal.
        v2f a0 = *(const v2f*)(wP + k);
        v2f a1 = *(const v2f*)(wP + k + 4);
        a0 *= keepA;
        a1 *= keepA;

        // D = A x B + C  (f32 WMMA: neg_a, A, neg_b, B, c_mod, C, reuse_a, reuse_b)
        acc0 = __builtin_amdgcn_wmma_f32_16x16x4_f32(false, a0, false, b0,
                                                     (short)0, acc0, false, false);
        acc1 = __builtin_amdgcn_wmma_f32_16x16x4_f32(false, a1, false, b1,
                                                     (short)0, acc1, false, false);

        // sum-of-squares rides along on already-loaded fea data (packed FMA)
        ss2 += b0 * b0;
        ss2 += b1 * b1;
    }

    v8f acc = acc0 + acc1;

    // horizontal add of the packed accumulator, then combine wave halves:
    // lane n covered K%4 in {0,1}, lane n+16 covered {2,3} of the same row.
    float ss = ss2.x + ss2.y;
    ss += __shfl_xor(ss, 16, 32);
    const float rn = 1.0f / sqrtf(ss);

    // D layout: VGPR c, lanes 0..15 => class c, fea row = lane. Lane r owns row
    // r's 4 logits -> one 16-byte store per output row.
    if (h == 0) {
        const float4 bv = *(const float4*)bias;
        float4 o;
        o.x = (acc[0] * rn + bv.x) * kScale;
        o.y = (acc[1] * rn + bv.y) * kScale;
        o.z = (acc[2] * rn + bv.z) * kScale;
        o.w = (acc[3] * rn + bv.w) * kScale;
        *(float4*)(out + (size_t)(row0 + n) * 4) = o;
    }
}

extern "C" void kernel_launch(void* const* d_in, const int* in_sizes, int n_in,
                              void* d_out, int out_size, void* d_ws, size_t ws_size,
                              hipStream_t stream) {
    const float* fea  = (const float*)d_in[0];   // [16384, 2048] f32
    const float* W    = (const float*)d_in[1];   // [4, 2048] f32
    const float* bias = (const float*)d_in[2];   // [4] f32
    float* out = (float*)d_out;                  // [16384, 4] f32

    const int numRows = in_sizes[0] / kEmb;
    const int tiles   = (numRows + kRowsPerWave - 1) / kRowsPerWave;
    const int blocks  = (tiles + kWavesPerBlock - 1) / kWavesPerBlock;

    binary_head_wmma_f32<<<blocks, kThreadsPerBlock, 0, stream>>>(
        fea, W, bias, out, numRows);
}